// YvChunkedParallelScan_72112500900643
// MI455X (gfx1250) — compile-verified
//
#include <hip/hip_runtime.h>

#define BATCH 2
#define SEQ   2048
#define DM    1024
#define NS    32
#define CHK   256
#define NC    (SEQ / CHK)   // 8 chunks
#define DT    128           // channels per scan workgroup

typedef float v2f __attribute__((ext_vector_type(2)));
typedef float v8f __attribute__((ext_vector_type(8)));

// ---------------------------------------------------------------------------
// Kernel 1: intra-chunk sequential scan (memory-bound part).
// One thread owns one channel d and keeps all 32 states in VGPRs.
// B/C tiles for the chunk (256 x 32 f32 each) are staged in LDS and read as
// uniform broadcasts inside the t-loop.
// Writes: out[b,s,d] = y_intra + u*D   and the chunk-final state S[b,c,n,d].
// ---------------------------------------------------------------------------
__global__ __launch_bounds__(DT) void scan_kernel(
    const float* __restrict__ u,  const float* __restrict__ Bm,
    const float* __restrict__ Cm, const float* __restrict__ Al,
    const float* __restrict__ Dp, float* __restrict__ out,
    float* __restrict__ S)
{
  __shared__ float Bs[CHK * NS];
  __shared__ float Cs[CHK * NS];

  const int bc = blockIdx.x;
  const int b  = bc / NC;
  const int c  = bc % NC;
  const int d  = blockIdx.y * DT + threadIdx.x;
  const size_t chunk_base = (size_t)b * SEQ + (size_t)c * CHK;

  // [t][n] is contiguous in Bmat/Cmat memory -> fully coalesced tile load.
  const float* Bg = Bm + chunk_base * NS;
  const float* Cg = Cm + chunk_base * NS;
  for (int i = threadIdx.x; i < CHK * NS; i += DT) {
    Bs[i] = Bg[i];
    Cs[i] = Cg[i];
  }

  float a[NS], h[NS];
#pragma unroll
  for (int n = 0; n < NS; ++n) {
    a[n] = -__expf(Al[n * DM + d]);   // A = -exp(A_log)
    h[n] = 0.0f;
  }
  const float dpar = Dp[d];
  __syncthreads();

  for (int t = 0; t < CHK; ++t) {
    const float ut = u[(chunk_base + t) * DM + d];
    float y = 0.0f;
#pragma unroll
    for (int n = 0; n < NS; ++n) {
      h[n] = a[n] * h[n] + Bs[t * NS + n] * ut;
      y   += Cs[t * NS + n] * h[n];
    }
    out[(chunk_base + t) * DM + d] = y + ut * dpar;
  }

#pragma unroll
  for (int n = 0; n < NS; ++n)
    S[(((size_t)b * NC + c) * NS + n) * DM + d] = h[n];
}

// ---------------------------------------------------------------------------
// Kernel 2: cross-chunk output correction via f32 WMMA.
//   Corr[t,d] = sum_n C[b, c*CHK+t, n] * ( -exp(A_log[n,d]) * S[b,c-1,n,d] )
//   out[b, c*CHK+t, d] += Corr[t,d]        (chunks c = 1..7)
// 256x32 @ 32x1024 GEMM per (b,c) built from V_WMMA_F32_16X16X4_F32
// (8 accumulating WMMAs per 16x16 tile, K = 32).
//
// ISA layouts used:
//  A (16x4 f32): lanes 0-15 -> {K=0,K=1}, lanes 16-31 -> {K=2,K=3}, M = lane&15
//  B (4x16 f32): lanes 0-15 -> {K=0,K=1}, lanes 16-31 -> {K=2,K=3}, N = lane&15
//  D (16x16 f32): VGPR i -> row i (lanes 0-15) / row i+8 (lanes 16-31)
// ---------------------------------------------------------------------------
__global__ __launch_bounds__(256) void corr_kernel(
    const float* __restrict__ Cm, const float* __restrict__ Al,
    const float* __restrict__ S,  float* __restrict__ out)
{
  const int bc    = blockIdx.x;        // 0 .. BATCH*(NC-1)-1
  const int b     = bc / (NC - 1);
  const int cprev = bc % (NC - 1);     // source chunk 0..6
  const int c     = cprev + 1;         // destination chunk 1..7
  const int wave  = threadIdx.x >> 5;  // 0..7
  const int lane  = threadIdx.x & 31;
  const int lanelo = lane & 15;
  const int hi    = lane >> 4;         // 0 or 1
  const int koff  = hi * 2;            // K pair offset per half-wave

  // A-operand: C rows for this wave's two 16-row tiles (reused over col tiles)
  v2f aop[2][8];
#pragma unroll
  for (int r = 0; r < 2; ++r) {
    const int t0 = (wave * 2 + r) * 16;
    const size_t row = (size_t)b * SEQ + (size_t)c * CHK + t0 + lanelo;
    const float* crow = Cm + row * NS;
#pragma unroll
    for (int k = 0; k < 8; ++k) {
      aop[r][k].x = crow[4 * k + koff];
      aop[r][k].y = crow[4 * k + koff + 1];
    }
  }

  const int dbase = blockIdx.y * 64;
  const size_t sbase = ((size_t)b * NC + cprev) * NS;

  for (int ct = 0; ct < 4; ++ct) {
    const int col = dbase + ct * 16 + lanelo;

    // B-operand: M[n, col] = -exp(A_log[n,col]) * S_prev[n,col]
    v2f bop[8];
#pragma unroll
    for (int k = 0; k < 8; ++k) {
      const int n0 = 4 * k + koff;
      bop[k].x = -__expf(Al[(size_t)n0 * DM + col]) *
                 S[(sbase + n0) * DM + col];
      bop[k].y = -__expf(Al[(size_t)(n0 + 1) * DM + col]) *
                 S[(sbase + n0 + 1) * DM + col];
    }

#pragma unroll
    for (int r = 0; r < 2; ++r) {
      v8f acc = {};
#pragma unroll
      for (int k = 0; k < 8; ++k) {
        acc = __builtin_amdgcn_wmma_f32_16x16x4_f32(
            false, aop[r][k], false, bop[k], (short)0, acc, false, false);
      }
      const int t0 = (wave * 2 + r) * 16 + hi * 8;
      const size_t obase =
          ((size_t)b * SEQ + (size_t)c * CHK + t0) * DM + col;
#pragma unroll
      for (int i = 0; i < 8; ++i)
        out[obase + (size_t)i * DM] += acc[i];
    }
  }
}

// ---------------------------------------------------------------------------
extern "C" void kernel_launch(void* const* d_in, const int* in_sizes, int n_in,
                              void* d_out, int out_size, void* d_ws, size_t ws_size,
                              hipStream_t stream) {
  const float* u  = (const float*)d_in[0];
  // d_in[1] = delta: accepted by the reference signature but unused.
  const float* Bm = (const float*)d_in[2];
  const float* Cm = (const float*)d_in[3];
  const float* Al = (const float*)d_in[4];
  const float* Dp = (const float*)d_in[5];
  float* out = (float*)d_out;
  float* S   = (float*)d_ws;   // BATCH*NC*NS*DM floats = 2 MB of chunk states

  dim3 g1(BATCH * NC, DM / DT);     // (16, 8) workgroups, 128 threads
  scan_kernel<<<g1, DT, 0, stream>>>(u, Bm, Cm, Al, Dp, out, S);

  dim3 g2(BATCH * (NC - 1), DM / 64);  // (14, 16) workgroups, 256 threads
  corr_kernel<<<g2, 256, 0, stream>>>(Cm, Al, S, out);
}